// BlockRelu_79972291051870
// MI455X (gfx1250) — compile-verified
//
#include <hip/hip_runtime.h>

#define HW   50176   // 224*224
#define WID  224

typedef int v4i __attribute__((vector_size(16)));
typedef float vf4 __attribute__((ext_vector_type(4)));
typedef __attribute__((address_space(1))) v4i* gptr_v4i;
typedef __attribute__((address_space(3))) v4i* lptr_v4i;

__device__ __forceinline__ float maskf(float s) {
  // (sign(s)+1)*0.5 :  s>0 -> 1,  s<0 -> 0,  s==0 -> 0.5
  return s > 0.0f ? 1.0f : (s < 0.0f ? 0.0f : 0.5f);
}

__device__ __forceinline__ float4 nt_load4(const float* p) {
  vf4 v = __builtin_nontemporal_load((const vf4*)p);
  return make_float4(v.x, v.y, v.z, v.w);
}
__device__ __forceinline__ void nt_store4(float* p, float4 f) {
  vf4 v = {f.x, f.y, f.z, f.w};
  __builtin_nontemporal_store(v, (vf4*)p);
}

// ---------------- 1x1 group (channels 0..15): mask*x == relu(x) ----------------
__global__ __launch_bounds__(256) void k_relu_1x1(const float* __restrict__ x,
                                                  float* __restrict__ y) {
  int t = blockIdx.x * 256 + threadIdx.x;      // float4 index over group
  int p = t / 12544;                           // plane 0..511   (HW/4 = 12544)
  int i = t - p * 12544;
  int b = p >> 4, c = p & 15;
  size_t off = (size_t)(b * 64 + c) * HW + (size_t)i * 4;
  float4 v = nt_load4(x + off);
  float4 o;
  o.x = fmaxf(v.x, 0.0f); o.y = fmaxf(v.y, 0.0f);
  o.z = fmaxf(v.z, 0.0f); o.w = fmaxf(v.w, 0.0f);
  nt_store4(y + off, o);
}

// ---------------- 2x2 group (channels 16..31): thread = 2 rows x 4 cols --------
__global__ __launch_bounds__(256) void k_pool_2x2(const float* __restrict__ x,
                                                  float* __restrict__ y) {
  int t   = blockIdx.x * 256 + threadIdx.x;
  int p   = t / 6272;                          // (112 row-pairs)*(56 col4) = 6272
  int rem = t - p * 6272;
  int r2  = rem / 56;
  int c4  = rem - r2 * 56;
  int b = p >> 4, c = 16 + (p & 15);
  size_t base = (size_t)(b * 64 + c) * HW + (size_t)(r2 * 2) * WID + (size_t)c4 * 4;
  float4 a = nt_load4(x + base);
  float4 d = nt_load4(x + base + WID);
  float m0 = maskf((a.x + a.y) + (d.x + d.y));
  float m1 = maskf((a.z + a.w) + (d.z + d.w));
  float4 oa = {a.x * m0, a.y * m0, a.z * m1, a.w * m1};
  float4 od = {d.x * m0, d.y * m0, d.z * m1, d.w * m1};
  nt_store4(y + base,       oa);
  nt_store4(y + base + WID, od);
}

// ---------------- 4x4 group (channels 32..47): thread = one 4x4 block ----------
__global__ __launch_bounds__(256) void k_pool_4x4(const float* __restrict__ x,
                                                  float* __restrict__ y) {
  int t   = blockIdx.x * 256 + threadIdx.x;
  int p   = t / 3136;                          // 56*56 blocks per plane
  int rem = t - p * 3136;
  int r   = rem / 56;
  int cc  = rem - r * 56;
  int b = p >> 4, c = 32 + (p & 15);
  size_t base = (size_t)(b * 64 + c) * HW + (size_t)(r * 4) * WID + (size_t)cc * 4;
  float4 v0 = nt_load4(x + base);
  float4 v1 = nt_load4(x + base + WID);
  float4 v2 = nt_load4(x + base + 2 * WID);
  float4 v3 = nt_load4(x + base + 3 * WID);
  float s = (((v0.x + v0.y) + (v0.z + v0.w)) + ((v1.x + v1.y) + (v1.z + v1.w)))
          + (((v2.x + v2.y) + (v2.z + v2.w)) + ((v3.x + v3.y) + (v3.z + v3.w)));
  float m = maskf(s);
  float4 o0 = {v0.x * m, v0.y * m, v0.z * m, v0.w * m};
  float4 o1 = {v1.x * m, v1.y * m, v1.z * m, v1.w * m};
  float4 o2 = {v2.x * m, v2.y * m, v2.z * m, v2.w * m};
  float4 o3 = {v3.x * m, v3.y * m, v3.z * m, v3.w * m};
  nt_store4(y + base,           o0);
  nt_store4(y + base + WID,     o1);
  nt_store4(y + base + 2 * WID, o2);
  nt_store4(y + base + 3 * WID, o3);
}

// ---------------- 7x7 group (channels 48..63): LDS-staged strip ----------------
// One workgroup handles a 7x224 strip (one block-row of one plane): 1568 floats.
// Staged into LDS via async global->LDS b128 DMA (ASYNCcnt).
__global__ __launch_bounds__(256) void k_pool_7x7(const float* __restrict__ x,
                                                  float* __restrict__ y) {
  __shared__ float tile[7 * WID];   // 1568 floats = 392 float4
  __shared__ float masks[32];

  int p  = blockIdx.x >> 5;         // plane 0..511
  int br = blockIdx.x & 31;         // block-row 0..31
  int b = p >> 4, c = 48 + (p & 15);
  size_t gbase = (size_t)(b * 64 + c) * HW + (size_t)(br * 7) * WID;
  int t = threadIdx.x;

  // Stage strip into LDS (each thread 1-2 b128 async DMA transfers; coalesced).
  for (int j = t; j < 392; j += 256) {
#if __has_builtin(__builtin_amdgcn_global_load_async_to_lds_b128)
    __builtin_amdgcn_global_load_async_to_lds_b128(
        (gptr_v4i)(x + gbase + (size_t)j * 4),
        (lptr_v4i)(&tile[j * 4]),
        0, 0);
#else
    *(float4*)(&tile[j * 4]) = *(const float4*)(x + gbase + (size_t)j * 4);
#endif
  }
#if __has_builtin(__builtin_amdgcn_s_wait_asynccnt)
  __builtin_amdgcn_s_wait_asynccnt(0);
#else
  asm volatile("s_wait_asynccnt 0" ::: "memory");
#endif
  __syncthreads();

  // 32 block sums per strip; lane t reads cols 7t..7t+6 (stride 7 is coprime
  // with 64 LDS banks -> conflict-free across the wave).
  if (t < 32) {
    float s = 0.0f;
    int col0 = t * 7;
#pragma unroll
    for (int r = 0; r < 7; ++r) {
      int rb = r * WID + col0;
      s += ((tile[rb] + tile[rb + 1]) + (tile[rb + 2] + tile[rb + 3]))
         + ((tile[rb + 4] + tile[rb + 5]) + tile[rb + 6]);
    }
    masks[t] = maskf(s);
  }
  __syncthreads();

  // Multiply & stream out with NT b128 stores (224 % 4 == 0 -> float4 in-row).
  for (int j = t; j < 392; j += 256) {
    float4 v = *(const float4*)(&tile[j * 4]);
    int f = j * 4;
    int col = f % WID;
    float4 o;
    o.x = v.x * masks[col / 7];
    o.y = v.y * masks[(col + 1) / 7];
    o.z = v.z * masks[(col + 2) / 7];
    o.w = v.w * masks[(col + 3) / 7];
    nt_store4(y + gbase + (size_t)f, o);
  }
}

extern "C" void kernel_launch(void* const* d_in, const int* in_sizes, int n_in,
                              void* d_out, int out_size, void* d_ws, size_t ws_size,
                              hipStream_t stream) {
  (void)in_sizes; (void)n_in; (void)out_size; (void)d_ws; (void)ws_size;
  const float* x = (const float*)d_in[0];
  float*       y = (float*)d_out;

  // Group sizes: 512 planes each (32 batches x 16 channels), HW = 50176.
  k_relu_1x1<<<dim3(25088), dim3(256), 0, stream>>>(x, y);   // 512*12544/256
  k_pool_2x2<<<dim3(12544), dim3(256), 0, stream>>>(x, y);   // 512*6272/256
  k_pool_4x4<<<dim3(6272),  dim3(256), 0, stream>>>(x, y);   // 512*3136/256
  k_pool_7x7<<<dim3(16384), dim3(256), 0, stream>>>(x, y);   // 512 planes * 32 strips
}